// Node_attention_layer_64029372449404
// MI455X (gfx1250) — compile-verified
//
#include <hip/hip_runtime.h>
#include <math.h>

// CDNA5 wave32 f32 WMMA fragments
typedef __attribute__((ext_vector_type(2))) float v2f;
typedef __attribute__((ext_vector_type(8))) float v8f;

// ---------------------------------------------------------------------------
// Generic per-batch GEMM:  C[b] = act( A[b] (MxK) @ B[b] (KxN) + bias )
// One wave computes a 16 x (16*NT) tile: NT accumulators share one A fragment
// per k-step (4x fewer A loads per WMMA).
// A frag (16x4, MxK): lane l (M=l&15), v0/v1 hold K = k0+2*(l>>4) +0/+1
// B frag (4x16, KxN): lane l (N=l&15), v0/v1 hold K = k0+2*(l>>4) +0/+1
// C/D (16x16): vgpr g, lane l -> row = g + 8*(l>>4), col = l&15
// ---------------------------------------------------------------------------
template <int NT>
__global__ __launch_bounds__(256) void gemm_f32_wmma(
    const float* __restrict__ A, const float* __restrict__ Bm,
    const float* __restrict__ bias,
    float* __restrict__ C, float* __restrict__ CT,
    int Kd, int lda, int ldb, int ldc, int ldct,
    long sA, long sB, long sC, long sCT,
    int mtiles, int ngroups, int act)
{
    const int gw   = (blockIdx.x * blockDim.x + threadIdx.x) >> 5;
    const int lane = threadIdx.x & 31;
    const int groupsPerBatch = mtiles * ngroups;
    const int b  = gw / groupsPerBatch;
    const int t  = gw % groupsPerBatch;
    const int mt = t / ngroups;
    const int ng = t % ngroups;            // column group of width 16*NT
    const int half = lane >> 4;
    const int l15  = lane & 15;

    const float* Ab = A  + (long)b * sA + (long)(mt * 16 + l15) * lda;
    const float* Bb = Bm + (long)b * sB + ng * (16 * NT) + l15;

    v8f acc[NT] = {};
    for (int k0 = 0; k0 < Kd; k0 += 4) {
        const int ka = k0 + 2 * half;
        const v2f av = *(const v2f*)(Ab + ka);   // 8B aligned (ka even, lda even)
#pragma unroll
        for (int j = 0; j < NT; ++j) {
            v2f bv;
            bv.x = Bb[(long)ka * ldb + j * 16];
            bv.y = Bb[(long)(ka + 1) * ldb + j * 16];
            acc[j] = __builtin_amdgcn_wmma_f32_16x16x4_f32(
                false, av, false, bv, (short)0, acc[j], false, false);
        }
    }

    float* Cb = C + (long)b * sC + (long)(mt * 16) * ldc + ng * (16 * NT) + l15;
#pragma unroll
    for (int j = 0; j < NT; ++j) {
        const float bias_v = bias ? bias[ng * (16 * NT) + j * 16 + l15] : 0.0f;
#pragma unroll
        for (int g = 0; g < 8; ++g) {
            const int r = g + 8 * half;
            float v = acc[j][g] + bias_v;
            if (act) v = tanhf(v);
            Cb[(long)r * ldc + j * 16] = v;
            if (CT)
                CT[(long)b * sCT + (long)(ng * (16 * NT) + j * 16 + l15) * ldct
                   + (mt * 16 + r)] = v;
        }
    }
}

// ---------------------------------------------------------------------------
// Gather k_emb^T with the 1/sqrt(D) scale folded in:
//   kembT[b][d][k] = emb_table[key_concepts[b][k]][d] * D^-0.5
// ---------------------------------------------------------------------------
__global__ __launch_bounds__(128) void kemb_gather(
    const int* __restrict__ keys, const float* __restrict__ emb,
    float* __restrict__ kembT)
{
    const int bk = blockIdx.x;             // B*K = 256 blocks
    const int b = bk >> 4, k = bk & 15;
    const long row = (long)keys[bk] * 512;
    const float scale = 0.044194173824159216f;   // 512^-0.5
    float* dst = kembT + (long)b * (512 * 16);
    for (int d = threadIdx.x; d < 512; d += blockDim.x)
        dst[d * 16 + k] = emb[row + d] * scale;
}

// ---------------------------------------------------------------------------
// Row softmax over N=64 (in place -> scores) and k_scores = max softmax over
// K=16.  One wave32 per (b,s) row.  Masked rows are exactly uniform.
// ---------------------------------------------------------------------------
__global__ __launch_bounds__(256) void softmax_kernel(
    float* __restrict__ logits,             // (B*S, 64) in/out
    const float* __restrict__ klogits,      // (B*S, 16)
    const unsigned char* __restrict__ mask, // (B*S) bool
    float* __restrict__ kscores)            // (B*S)
{
    const int row  = (blockIdx.x * blockDim.x + threadIdx.x) >> 5;
    const int lane = threadIdx.x & 31;
    float* Lr = logits + (long)row * 64;

    if (mask[row]) {                        // whole row NEG -> uniform softmax
        Lr[lane]      = 1.0f / 64.0f;
        Lr[lane + 32] = 1.0f / 64.0f;
        if (lane == 0) kscores[row] = 1.0f / 16.0f;
        return;
    }

    float v0 = Lr[lane], v1 = Lr[lane + 32];
    float m = fmaxf(v0, v1);
    for (int off = 16; off; off >>= 1) m = fmaxf(m, __shfl_xor(m, off));
    float e0 = expf(v0 - m), e1 = expf(v1 - m);
    float s = e0 + e1;
    for (int off = 16; off; off >>= 1) s += __shfl_xor(s, off);
    const float inv = 1.0f / s;
    Lr[lane]      = e0 * inv;
    Lr[lane + 32] = e1 * inv;

    float kv = (lane < 16) ? klogits[(long)row * 16 + lane] : -3.0e38f;
    float km = kv;
    for (int off = 16; off; off >>= 1) km = fmaxf(km, __shfl_xor(km, off));
    float ke = (lane < 16) ? expf(kv - km) : 0.0f;
    float ks = ke;
    for (int off = 16; off; off >>= 1) ks += __shfl_xor(ks, off);
    float p = ke / ks;                      // lanes >=16 contribute 0
    for (int off = 16; off; off >>= 1) p = fmaxf(p, __shfl_xor(p, off));
    if (lane == 0) kscores[row] = p;
}

// ---------------------------------------------------------------------------
// Fused output GEMM (16x64 tile per wave):
//   out[b] = tanh( enc[b](2048x512) @ W_hid_top(512x512)
//                + scores[b](2048x64) @ P2[b](64x512) + b_hid )
// ---------------------------------------------------------------------------
__global__ __launch_bounds__(256) void out_gemm_wmma(
    const float* __restrict__ enc, const float* __restrict__ Whid,
    const float* __restrict__ scores, const float* __restrict__ P2,
    const float* __restrict__ bhid, float* __restrict__ out)
{
    const int gw   = (blockIdx.x * blockDim.x + threadIdx.x) >> 5;
    const int lane = threadIdx.x & 31;
    const int b  = gw >> 10;                // 128 mtiles * 8 ngroups per batch
    const int t  = gw & 1023;
    const int mt = t >> 3;                  // 0..127
    const int ng = t & 7;                   // 0..7  (64-wide column group)
    const int half = lane >> 4;
    const int l15  = lane & 15;

    v8f acc[4] = {};

    // Phase 1: enc @ W_hid rows [0,512)
    const float* Ab = enc + ((long)b * 2048 + mt * 16 + l15) * 512;
    const float* Bp = Whid + ng * 64 + l15;
    for (int k0 = 0; k0 < 512; k0 += 4) {
        const int ka = k0 + 2 * half;
        __builtin_prefetch(Bp + (long)(ka + 32) * 512, 0, 1);  // global_prefetch_b8
        const v2f av = *(const v2f*)(Ab + ka);
#pragma unroll
        for (int j = 0; j < 4; ++j) {
            v2f bv;
            bv.x = Bp[(long)ka * 512 + j * 16];
            bv.y = Bp[(long)(ka + 1) * 512 + j * 16];
            acc[j] = __builtin_amdgcn_wmma_f32_16x16x4_f32(
                false, av, false, bv, (short)0, acc[j], false, false);
        }
    }

    // Phase 2: scores @ P2[b]   (== dot_x @ W_hid rows [512,1024))
    const float* A2 = scores + ((long)b * 2048 + mt * 16 + l15) * 64;
    const float* B2 = P2 + (long)b * (64 * 512) + ng * 64 + l15;
    for (int k0 = 0; k0 < 64; k0 += 4) {
        const int ka = k0 + 2 * half;
        const v2f av = *(const v2f*)(A2 + ka);
#pragma unroll
        for (int j = 0; j < 4; ++j) {
            v2f bv;
            bv.x = B2[(long)ka * 512 + j * 16];
            bv.y = B2[(long)(ka + 1) * 512 + j * 16];
            acc[j] = __builtin_amdgcn_wmma_f32_16x16x4_f32(
                false, av, false, bv, (short)0, acc[j], false, false);
        }
    }

    float* Ob = out + ((long)b * 2048 + mt * 16) * 512 + ng * 64 + l15;
#pragma unroll
    for (int j = 0; j < 4; ++j) {
        const float bias_v = bhid[ng * 64 + j * 16 + l15];
#pragma unroll
        for (int g = 0; g < 8; ++g) {
            const int r = g + 8 * half;
            Ob[(long)r * 512 + j * 16] = tanhf(acc[j][g] + bias_v);
        }
    }
}

// ---------------------------------------------------------------------------
extern "C" void kernel_launch(void* const* d_in, const int* in_sizes, int n_in,
                              void* d_out, int out_size, void* d_ws, size_t ws_size,
                              hipStream_t stream)
{
    const float* enc   = (const float*)d_in[0];          // (16,2048,512)
    const float* x     = (const float*)d_in[1];          // (16,64,1024)
    const int*   keys  = (const int*)d_in[2];            // (16,16)
    const unsigned char* mask = (const unsigned char*)d_in[3]; // (16,2048) bool
    const float* W_att = (const float*)d_in[4];          // (1024,512)
    const float* b_att = (const float*)d_in[5];          // (512)
    const float* W_hid = (const float*)d_in[6];          // (1024,512)
    const float* b_hid = (const float*)d_in[7];          // (512)
    const float* emb   = (const float*)d_in[8];          // (32000,512)

    float* out     = (float*)d_out;                      // (16,2048,512)
    float* kscores = out + 16L * 2048 * 512;             // (16,2048)

    float* ws        = (float*)d_ws;
    float* projected = ws;                   // 16*64*512   = 524288
    float* projT     = projected + 524288;   // 16*512*64   = 524288
    float* P2        = projT + 524288;       // 16*64*512   = 524288
    float* kembT     = P2 + 524288;          // 16*512*16   = 131072
    float* logits    = kembT + 131072;       // 16*2048*64  = 2097152 (-> scores)
    float* klogits   = logits + 2097152;     // 16*2048*16  = 524288
    // total ws: 4,325,376 floats = 17.3 MB

    // 1) gather scaled k_emb^T
    kemb_gather<<<256, 128, 0, stream>>>(keys, emb, kembT);

    // 2) projected = tanh(x @ W_att + b_att), plus projected^T
    //    per batch (64x1024)@(1024x512): 4 mtiles x 8 groups -> 512 waves
    gemm_f32_wmma<4><<<512 / 8, 256, 0, stream>>>(
        x, W_att, b_att, projected, projT,
        1024, 1024, 512, 512, 64,
        64L * 1024, 0, 64L * 512, 512L * 64,
        4, 8, 1);

    // 3) P2 = projected @ W_hid[512:,:]   (64x512)@(512x512) -> 512 waves
    gemm_f32_wmma<4><<<512 / 8, 256, 0, stream>>>(
        projected, W_hid + 512L * 512, nullptr, P2, nullptr,
        512, 512, 512, 512, 0,
        64L * 512, 0, 64L * 512, 0,
        4, 8, 0);

    // 4) logits = enc @ projected^T  (2048x512)@(512x64): 128x1 groups -> 2048 waves
    gemm_f32_wmma<4><<<2048 / 8, 256, 0, stream>>>(
        enc, projT, nullptr, logits, nullptr,
        512, 512, 64, 64, 0,
        2048L * 512, 512L * 64, 2048L * 64, 0,
        128, 1, 0);

    // 5) klogits = (scale*enc) @ k_emb^T  (2048x512)@(512x16): N=16 -> NT=1
    gemm_f32_wmma<1><<<2048 / 8, 256, 0, stream>>>(
        enc, kembT, nullptr, klogits, nullptr,
        512, 512, 16, 16, 0,
        2048L * 512, 512L * 16, 2048L * 16, 0,
        128, 1, 0);

    // 6) masked softmax (in place -> scores) + k_scores ; 32768 rows, 1 wave each
    softmax_kernel<<<4096, 256, 0, stream>>>(logits, klogits, mask, kscores);

    // 7) out = tanh(enc@W_hid_top + scores@P2 + b_hid) ; 16384 waves
    out_gemm_wmma<<<16384 / 8, 256, 0, stream>>>(enc, W_hid, logits, P2, b_hid, out);
}